// RealSHT_11733850653276
// MI455X (gfx1250) — compile-verified
//
#include <hip/hip_runtime.h>

// ---- problem constants -----------------------------------------------------
#define NLAT   721
#define NLON   1440
#define LMAX   360
#define MMAX   360
#define NB     16                 // 2*8 flattened batch
#define NF2    (2*MMAX)           // 720: [cos columns | -sin columns]
#define KQ     181                // ceil(721/4): K-groups of 4 in stage 2
#define PLANE  ((size_t)MMAX*KQ*NB*4)   // floats per re/im plane = 4,170,240
#define PI_F   3.14159265358979323846f

typedef __attribute__((ext_vector_type(2))) float v2f;
typedef __attribute__((ext_vector_type(8))) float v8f;

// ---------------------------------------------------------------------------
// Stage 0 (prep):
//  a) trig table in k-paired layout: T2[kp][j][bit] = dft(n = 2*kp+bit, j)
//     j <  360:  cos(2*pi*j*n/1440) * (2*pi/1440)
//     j >= 360: -sin(2*pi*(j-360)*n/1440) * (2*pi/1440)
//     -> stage-1 B fragment is one aligned b64 load.
//  b) zero the K-pad (k = 721..723) of the XF tensor so stage 2 needs no
//     guards: XF[plane][m][kq=180][b][kr=1..3] = 0.
// ---------------------------------------------------------------------------
__global__ void sht_prep_kernel(float* __restrict__ T2, float* __restrict__ XF) {
    int tid = blockIdx.x * blockDim.x + threadIdx.x;
    if (tid >= NLON * NF2) return;

    // --- trig table ---
    int bit = tid & 1;
    int t2  = tid >> 1;
    int j   = t2 % NF2;
    int kp  = t2 / NF2;
    int n   = kp * 2 + bit;
    int m   = (j < MMAX) ? j : (j - MMAX);
    int p   = (m * n) % NLON;                      // exact angle reduction
    float ang   = (2.0f * PI_F / (float)NLON) * (float)p;
    float scale =  2.0f * PI_F / (float)NLON;
    float v = (j < MMAX) ? __cosf(ang) : -__sinf(ang);
    T2[tid] = v * scale;

    // --- zero XF K-pad: 2 planes * 360 m * 16 b * 3 kr = 34560 ---
    if (tid < 2 * MMAX * NB * 3) {
        int pl   = tid / (MMAX * NB * 3);
        int rem  = tid - pl * (MMAX * NB * 3);
        int mm   = rem / (NB * 3);
        int rem2 = rem - mm * (NB * 3);
        int b    = rem2 / 3;
        int kr   = 1 + rem2 - b * 3;
        size_t idx = (size_t)pl * PLANE + (((size_t)mm * KQ + 180) * NB + b) * 4 + kr;
        XF[idx] = 0.0f;
    }
}

// ---------------------------------------------------------------------------
// Stage 1: truncated rFFT as fp32 WMMA GEMM.
//   X : 11536 x 1440 (row r = b*721 + k_lat),  T2: paired-k DFT matrix.
//   Output scattered into WMMA-native layout:
//     XF[plane][m][kq][b][kr],  plane 0 = Re, 1 = Im,  k = kq*4 + kr.
// 11536 = 721*16 M-tiles exact, 720 = 45*16 N-tiles exact, K = 1440 = 360*4.
// Inner loop: 1 b64 A-load + 1 b64 B-load + 1 WMMA, no guards.
// ---------------------------------------------------------------------------
__global__ void __launch_bounds__(128)
sht_dft_kernel(const float* __restrict__ X,
               const float* __restrict__ T2,
               float*       __restrict__ XF) {
    const int lane = threadIdx.x & 31;
    const int wv   = threadIdx.x >> 5;
    const int nt   = blockIdx.x * 4 + wv;          // N tile: 0..44
    const int mt   = blockIdx.y;                   // M tile: 0..720
    if (nt >= 45) return;

    const int lo  = lane & 15;
    const int hi  = lane >> 4;                     // 0 -> K {0,1}, 1 -> K {2,3}
    const int row = mt * 16 + lo;                  // A M-index
    const int col = nt * 16 + lo;                  // B N-index
    const float* __restrict__ Arow = X + (size_t)row * NLON;

    v8f c = {};
    for (int k0 = 0; k0 < NLON; k0 += 4) {
        const int kk = k0 + hi * 2;                                    // even
        v2f a = *(const v2f*)(Arow + kk);                              // A[row][kk..kk+1]
        v2f b = *(const v2f*)(T2 + ((size_t)(kk >> 1) * NF2 + col) * 2); // B[kk..kk+1][col]
        c = __builtin_amdgcn_wmma_f32_16x16x4_f32(
                false, a, false, b, (short)0, c, false, false);
    }

    // scatter C tile into XF[plane][m][kq][b][kr]
    const int mcol  = (col < MMAX) ? col : (col - MMAX);
    const int plane = (col < MMAX) ? 0 : 1;
    float* __restrict__ Op = XF + (size_t)plane * PLANE;
    #pragma unroll
    for (int v = 0; v < 8; ++v) {
        int r  = mt * 16 + v + hi * 8;             // C/D: M = v + 8*hi
        int b  = (unsigned)r / NLAT;
        int k  = r - b * NLAT;
        int kq = (unsigned)k >> 2;
        int kr = k & 3;
        Op[(((size_t)mcol * KQ + kq) * NB + b) * 4 + kr] = c[v];
    }
}

// ---------------------------------------------------------------------------
// Stage 2: per-m Legendre contraction (fp32 WMMA), re+im share the B fragment.
//   A_re/A_im: one aligned b64 load each from WMMA-native XF layout.
//   B[k][l] = weights[m][l][k]: read-once 374 MB -> non-temporal b32 loads.
//   Main loop 180 steps guard-free; single guarded tail step for k=720.
//   OUT[b][l][m] complex interleaved (float2 stores).
// ---------------------------------------------------------------------------
__global__ void __launch_bounds__(128)
sht_legendre_kernel(const float* __restrict__ XF,
                    const float* __restrict__ W,
                    float*       __restrict__ OUT) {
    const int lane = threadIdx.x & 31;
    const int wv   = threadIdx.x >> 5;
    const int lt   = blockIdx.x * 4 + wv;          // l tile: 0..22
    const int m    = blockIdx.y;                   // 0..359
    if (lt >= 23) return;

    const int lo  = lane & 15;
    const int hi  = lane >> 4;
    const int l   = lt * 16 + lo;                  // B N-index
    const bool lok = (l < LMAX);
    const float* __restrict__ Wm =                 // clamp for memory safety;
        W + (size_t)m * LMAX * NLAT + (size_t)(lok ? l : 0) * NLAT;  // cols discarded at store

    const float* __restrict__ Ar =                 // lane's A base (re plane)
        XF + (((size_t)m * KQ) * NB + lo) * 4 + hi * 2;
    const float* __restrict__ Ai = Ar + PLANE;

    v8f cre = {};
    v8f cim = {};
    for (int ks = 0; ks < 180; ++ks) {             // guard-free main loop
        const int kk = ks * 4 + hi * 2;
        v2f ar = *(const v2f*)(Ar + (size_t)ks * (NB * 4));
        v2f ai = *(const v2f*)(Ai + (size_t)ks * (NB * 4));
        v2f bb;
        bb.x = __builtin_nontemporal_load(Wm + kk);
        bb.y = __builtin_nontemporal_load(Wm + kk + 1);
        cre = __builtin_amdgcn_wmma_f32_16x16x4_f32(
                  false, ar, false, bb, (short)0, cre, false, false);
        cim = __builtin_amdgcn_wmma_f32_16x16x4_f32(
                  false, ai, false, bb, (short)0, cim, false, false);
    }
    {   // tail: k0 = 720 (A pad pre-zeroed, weights guarded)
        v2f ar = *(const v2f*)(Ar + (size_t)180 * (NB * 4));
        v2f ai = *(const v2f*)(Ai + (size_t)180 * (NB * 4));
        v2f bb;
        bb.x = (hi == 0) ? __builtin_nontemporal_load(Wm + 720) : 0.0f;
        bb.y = 0.0f;
        cre = __builtin_amdgcn_wmma_f32_16x16x4_f32(
                  false, ar, false, bb, (short)0, cre, false, false);
        cim = __builtin_amdgcn_wmma_f32_16x16x4_f32(
                  false, ai, false, bb, (short)0, cim, false, false);
    }

    if (lok) {
        #pragma unroll
        for (int v = 0; v < 8; ++v) {
            int b = v + hi * 8;                                // C/D: M = v + 8*hi
            size_t idx = ((size_t)b * LMAX + l) * MMAX + m;    // complex index
            ((float2*)OUT)[idx] = make_float2(cre[v], cim[v]);
        }
    }
}

// ---------------------------------------------------------------------------
extern "C" void kernel_launch(void* const* d_in, const int* in_sizes, int n_in,
                              void* d_out, int out_size, void* d_ws, size_t ws_size,
                              hipStream_t stream) {
    const float* x = (const float*)d_in[0];   // (2,8,721,1440) f32
    const float* w = (const float*)d_in[1];   // (360,360,721)  f32
    float* out = (float*)d_out;               // complex64 interleaved

    float* T2 = (float*)d_ws;                 // 1440*720 f32         (4.15 MB)
    float* XF = T2 + (size_t)NLON * NF2;      // 2 planes * 4,170,240 (33.4 MB)

    // Stage 0: trig table + XF K-pad zeroing
    {
        int total = NLON * NF2;               // 1,036,800
        dim3 blk(256), grd((total + 255) / 256);
        sht_prep_kernel<<<grd, blk, 0, stream>>>(T2, XF);
    }
    // Stage 1: truncated rFFT as WMMA GEMM
    {
        dim3 blk(128), grd(12, 721);          // 4 waves/block * 12 = 48 >= 45 N-tiles
        sht_dft_kernel<<<grd, blk, 0, stream>>>(x, T2, XF);
    }
    // Stage 2: per-m Legendre WMMA GEMM
    {
        dim3 blk(128), grd(6, 360);           // 4 waves/block * 6 = 24 >= 23 l-tiles
        sht_legendre_kernel<<<grd, blk, 0, stream>>>(XF, w, out);
    }
}